// FeatModel_50611894616409
// MI455X (gfx1250) — compile-verified
//
#include <hip/hip_runtime.h>

// ---------------------------------------------------------------------------
// GAT pipeline for MI455X (gfx1250), wave32, fp32 WMMA + async-to-LDS staging.
// Shapes: B=16,P=256,C=768 -> NUM_L=4096 left nodes, NN=8192 nodes, H=12, D=64
// ---------------------------------------------------------------------------

typedef __attribute__((ext_vector_type(2))) float v2f;
typedef __attribute__((ext_vector_type(8))) float v8f;

#define CDIM 768
#define HEADS 12
#define HDIM 64
#define NUM_L 4096
#define NNODES 8192
#define KTOP 8
#define E1 (NUM_L * KTOP + NNODES)   // 40960
#define E2 (NNODES + NNODES)         // 16384
#define NEG_SLOPE 0.2f

// GEMM blocking
#define BM 64
#define BN 64
#define BK 16
#define APAD (BK + 4)   // 20 floats/row: 16B aligned, conflict-free column reads
#define BPAD (BN + 4)   // 68 floats/row: 16B aligned

// ---------------------------------------------------------------------------
// CDNA5 async global->LDS copy (ASYNCcnt path) [CDNA5 ISA]
// ---------------------------------------------------------------------------
__device__ __forceinline__ void async_copy_f4(const void* lds_dst, const float* g) {
  const unsigned ldsOff = (unsigned)(uintptr_t)lds_dst;   // flat->LDS: addr[31:0]
  asm volatile("global_load_async_to_lds_b128 %0, %1, off"
               :: "v"(ldsOff), "v"((unsigned long long)(uintptr_t)g)
               : "memory");
}
__device__ __forceinline__ void async_wait0() {
  asm volatile("s_wait_asynccnt 0x0" ::: "memory");
}

// ---------------------------------------------------------------------------
// Blocked fp32 GEMM: C[M,N] = A[M,K] @ B[K,N].
// 256 threads = 8 waves. Workgroup tile 64x64; wave (wm,wn) owns 16 rows x 32
// cols (two 16x16 WMMA accumulators). A/B chunks staged in LDS via async copy.
// fp32 A 16x4 lane layout: lanes 0-15 carry K={k,k+1} of row M=lane; lanes
// 16-31 carry K={k+2,k+3}. C/D: VGPR r -> M=r (lo half) / M=r+8 (hi half).
// ---------------------------------------------------------------------------
__global__ __launch_bounds__(256) void gemm_wmma_f32(const float* __restrict__ A,
                                                     const float* __restrict__ B,
                                                     float* __restrict__ C,
                                                     int Kdim, int N) {
  __shared__ float sA[BM][APAD];
  __shared__ float sB[BK][BPAD];
  const int tid = threadIdx.x;
  const int lane = tid & 31;
  const int wave = tid >> 5;      // 0..7
  const int wm = wave >> 1;       // 0..3 -> row sub-tile
  const int wn = wave & 1;        // 0..1 -> col sub-tile
  const int half = lane >> 4;
  const int l = lane & 15;
  const int rowBase = blockIdx.y * BM;
  const int colBase = blockIdx.x * BN;

  // staging maps: A: 64 rows x 4 float4-segments; B: 16 rows x 16 segments
  const int arow = tid >> 2, aseg = (tid & 3) << 2;
  const int brow = tid >> 4, bseg = (tid & 15) << 2;

  v8f c0 = {}, c1 = {};
  for (int k0 = 0; k0 < Kdim; k0 += BK) {
    async_copy_f4(&sA[arow][aseg], &A[(size_t)(rowBase + arow) * Kdim + k0 + aseg]);
    async_copy_f4(&sB[brow][bseg], &B[(size_t)(k0 + brow) * N + colBase + bseg]);
    async_wait0();
    __syncthreads();
#pragma unroll
    for (int kk = 0; kk < BK; kk += 4) {
      const int ka = kk + 2 * half;
      v2f a, b0, b1;
      a.x = sA[wm * 16 + l][ka];
      a.y = sA[wm * 16 + l][ka + 1];
      b0.x = sB[ka][wn * 32 + l];
      b0.y = sB[ka + 1][wn * 32 + l];
      b1.x = sB[ka][wn * 32 + 16 + l];
      b1.y = sB[ka + 1][wn * 32 + 16 + l];
      c0 = __builtin_amdgcn_wmma_f32_16x16x4_f32(false, a, false, b0, (short)0, c0, false, false);
      c1 = __builtin_amdgcn_wmma_f32_16x16x4_f32(false, a, false, b1, (short)0, c1, false, false);
    }
    __syncthreads();
  }
#pragma unroll
  for (int r = 0; r < 8; ++r) {
    const size_t ro = (size_t)(rowBase + wm * 16 + r + 8 * half) * N + colBase + wn * 32;
    C[ro + l] = c0[r];
    C[ro + 16 + l] = c1[r];
  }
}

// ---------------------------------------------------------------------------
// sim = ZL @ ZR^T with fused per-row top-8 (indices only).
// One wave per 16-row strip. The 16 ZL rows (48KB) are staged in LDS once via
// async copies and reused across all 256 column tiles; ZR is streamed from
// global (contiguous b64 per lane per WMMA step). Each 16x16 tile is spilled
// to LDS, then lanes 0-15 maintain a sorted top-8 (val,idx) list per row.
// ---------------------------------------------------------------------------
#define LPAD (CDIM + 4)   // 772: 16B aligned rows, bank stride 4
__global__ __launch_bounds__(32) void sim_topk_kernel(const float* __restrict__ ZL,
                                                      const float* __restrict__ ZR,
                                                      int* __restrict__ IDX) {
  __shared__ float sL[16][LPAD];
  __shared__ float tile[16][16];
  const int lane = threadIdx.x;
  const int half = lane >> 4;
  const int l = lane & 15;
  const int row = blockIdx.x * 16 + l;

  // Stage 16 rows of ZL: 16*768/4 = 3072 float4 chunks across 32 lanes.
  for (int i = lane; i < 16 * (CDIM / 4); i += 32) {
    const int r = i / (CDIM / 4);
    const int s = (i % (CDIM / 4)) << 2;
    async_copy_f4(&sL[r][s], &ZL[(size_t)(blockIdx.x * 16 + r) * CDIM + s]);
  }
  async_wait0();
  __syncthreads();

  float tv[KTOP];
  int ti[KTOP];
#pragma unroll
  for (int q = 0; q < KTOP; ++q) { tv[q] = -3.402823466e38f; ti[q] = 0; }

  for (int tj = 0; tj < NUM_L / 16; ++tj) {
    const int col = tj * 16 + l;
    v8f c = {};
    for (int k0 = 0; k0 < CDIM; k0 += 4) {
      const int ka = k0 + 2 * half;
      v2f a, b;
      a.x = sL[l][ka];
      a.y = sL[l][ka + 1];
      b.x = ZR[(size_t)col * CDIM + ka];      // B[k][n] = ZR[n][k]
      b.y = ZR[(size_t)col * CDIM + ka + 1];
      c = __builtin_amdgcn_wmma_f32_16x16x4_f32(false, a, false, b, (short)0, c, false, false);
    }
#pragma unroll
    for (int r = 0; r < 8; ++r) tile[r + 8 * half][l] = c[r];
    __syncthreads();
    if (lane < 16) {
#pragma unroll
      for (int j = 0; j < 16; ++j) {
        const float val = tile[l][j];
        if (val > tv[KTOP - 1]) {
          tv[KTOP - 1] = val;
          ti[KTOP - 1] = tj * 16 + j;
#pragma unroll
          for (int q = KTOP - 2; q >= 0; --q) {
            if (tv[q + 1] > tv[q]) {
              float fv = tv[q]; tv[q] = tv[q + 1]; tv[q + 1] = fv;
              int iv = ti[q]; ti[q] = ti[q + 1]; ti[q + 1] = iv;
            }
          }
        }
      }
    }
    __syncthreads();
  }
  if (lane < 16) {
#pragma unroll
    for (int q = 0; q < KTOP; ++q) IDX[row * KTOP + q] = ti[q];
  }
}

// ---------------------------------------------------------------------------
// Small helper kernels
// ---------------------------------------------------------------------------
__global__ void fill_kernel(float* __restrict__ p, float v, int n) {
  int t = blockIdx.x * blockDim.x + threadIdx.x;
  if (t < n) p[t] = v;
}

__global__ void concat_kernel(const float* __restrict__ L,
                              const float* __restrict__ R,
                              float* __restrict__ X) {
  int t = blockIdx.x * blockDim.x + threadIdx.x;
  const int halfn = NUM_L * CDIM;
  if (t < halfn) X[t] = L[t];
  else if (t < 2 * halfn) X[t] = R[t - halfn];
}

__global__ void build_edges1(const int* __restrict__ IDX, int* __restrict__ SRC,
                             int* __restrict__ DST) {
  int t = blockIdx.x * blockDim.x + threadIdx.x;
  if (t >= E1) return;
  if (t < NUM_L * KTOP) {
    SRC[t] = t / KTOP;
    DST[t] = NUM_L + IDX[t];
  } else {
    int n = t - NUM_L * KTOP;
    SRC[t] = n;
    DST[t] = n;
  }
}

__global__ void build_edges2(int* __restrict__ SRC, int* __restrict__ DST) {
  int t = blockIdx.x * blockDim.x + threadIdx.x;
  if (t >= E2) return;
  if (t < NNODES) {
    SRC[t] = t;
    DST[t] = (t / 256) * 256;   // image-representative node
  } else {
    int n = t - NNODES;
    SRC[t] = n;
    DST[t] = n;
  }
}

// el[n,h] = z[n,h,:] . a_src[h,:] ; er likewise
__global__ void attn_scores(const float* __restrict__ Z,
                            const float* __restrict__ a_s,
                            const float* __restrict__ a_d,
                            float* __restrict__ EL, float* __restrict__ ER) {
  int t = blockIdx.x * blockDim.x + threadIdx.x;
  if (t >= NNODES * HEADS) return;
  const int n = t / HEADS, h = t % HEADS;
  const float* zp = Z + (size_t)n * CDIM + h * HDIM;
  float sl = 0.f, sr = 0.f;
#pragma unroll 8
  for (int d = 0; d < HDIM; ++d) {
    const float zv = zp[d];
    sl += zv * a_s[h * HDIM + d];
    sr += zv * a_d[h * HDIM + d];
  }
  EL[t] = sl;
  ER[t] = sr;
}

__device__ inline void atomicMaxFloat(float* addr, float val) {
  unsigned* ua = (unsigned*)addr;
  unsigned old = *ua;
  while (__uint_as_float(old) < val) {
    unsigned prev = atomicCAS(ua, old, __float_as_uint(val));
    if (prev == old) break;
    old = prev;
  }
}

// e = leaky_relu(el[src]+er[dst]); segment-max into MX[dst]
__global__ void edge_logits_max(const int* __restrict__ SRC,
                                const int* __restrict__ DST,
                                const float* __restrict__ EL,
                                const float* __restrict__ ER,
                                float* __restrict__ EV, float* __restrict__ MX,
                                int E) {
  int t = blockIdx.x * blockDim.x + threadIdx.x;
  if (t >= E * HEADS) return;
  const int e = t / HEADS, h = t % HEADS;
  float x = EL[SRC[e] * HEADS + h] + ER[DST[e] * HEADS + h];
  x = (x > 0.f) ? x : NEG_SLOPE * x;
  EV[t] = x;
  atomicMaxFloat(&MX[DST[e] * HEADS + h], x);
}

// ex = exp(e - m[dst]); segment-sum into SX[dst]
__global__ void edge_exp_sum(const int* __restrict__ DST,
                             float* __restrict__ EV,
                             const float* __restrict__ MX,
                             float* __restrict__ SX, int E) {
  int t = blockIdx.x * blockDim.x + threadIdx.x;
  if (t >= E * HEADS) return;
  const int e = t / HEADS, h = t % HEADS;
  const float ex = __expf(EV[t] - MX[DST[e] * HEADS + h]);
  EV[t] = ex;
  atomicAdd(&SX[DST[e] * HEADS + h], ex);
}

// Y[dst] += (ex/s[dst]) * z[src], per head
__global__ void edge_scatter(const int* __restrict__ SRC,
                             const int* __restrict__ DST,
                             const float* __restrict__ EV,
                             const float* __restrict__ SX,
                             const float* __restrict__ Z,
                             float* __restrict__ Y, int E) {
  int t = blockIdx.x * blockDim.x + threadIdx.x;
  if (t >= E * HEADS) return;
  const int e = t / HEADS, h = t % HEADS;
  const float alpha = EV[t] / SX[DST[e] * HEADS + h];
  const float* zp = Z + (size_t)SRC[e] * CDIM + h * HDIM;
  float* yp = Y + (size_t)DST[e] * CDIM + h * HDIM;
#pragma unroll 8
  for (int d = 0; d < HDIM; ++d) atomicAdd(&yp[d], alpha * zp[d]);
}

// X = act(Y + b)  (act = ELU for inner layers, identity for last)
__global__ void finalize_kernel(const float* __restrict__ Y,
                                const float* __restrict__ bias,
                                float* __restrict__ X, int do_elu) {
  int t = blockIdx.x * blockDim.x + threadIdx.x;
  if (t >= NNODES * CDIM) return;
  float v = Y[t] + bias[t % CDIM];
  if (do_elu) v = (v > 0.f) ? v : (__expf(v) - 1.f);
  X[t] = v;
}

__global__ void gather_out(const float* __restrict__ X, float* __restrict__ out) {
  int t = blockIdx.x * blockDim.x + threadIdx.x;
  if (t >= 32 * CDIM) return;
  const int j = t / CDIM, c = t % CDIM;
  out[t] = X[(size_t)(j * 256) * CDIM + c];   // img_node[j] = j*256
}

// ---------------------------------------------------------------------------
// Host orchestration
// ---------------------------------------------------------------------------
extern "C" void kernel_launch(void* const* d_in, const int* in_sizes, int n_in,
                              void* d_out, int out_size, void* d_ws, size_t ws_size,
                              hipStream_t stream) {
  const float* l_feat = (const float*)d_in[0];   // [16,256,768] == [4096,768]
  const float* r_feat = (const float*)d_in[1];
  const float* Wg     = (const float*)d_in[2];   // [768,768]
  const float* Ws     = (const float*)d_in[3];   // [4,768,768]
  const float* a_src  = (const float*)d_in[4];   // [4,12,64]
  const float* a_dst  = (const float*)d_in[5];
  const float* bias   = (const float*)d_in[6];   // [4,768]
  float* out = (float*)d_out;

  // Workspace layout (256B aligned slabs)
  char* base = (char*)d_ws;
  size_t off = 0;
  auto grab = [&](size_t bytes) -> char* {
    char* p = base + off;
    off += (bytes + 255) & ~(size_t)255;
    return p;
  };
  float* X  = (float*)grab((size_t)NNODES * CDIM * 4);
  float* Z  = (float*)grab((size_t)NNODES * CDIM * 4);
  float* Y  = (float*)grab((size_t)NNODES * CDIM * 4);
  float* EL = (float*)grab((size_t)NNODES * HEADS * 4);
  float* ER = (float*)grab((size_t)NNODES * HEADS * 4);
  float* MX = (float*)grab((size_t)NNODES * HEADS * 4);
  float* SX = (float*)grab((size_t)NNODES * HEADS * 4);
  float* EV = (float*)grab((size_t)E1 * HEADS * 4);
  int* SRC  = (int*)grab((size_t)E1 * 4);
  int* DST  = (int*)grab((size_t)E1 * 4);
  int* IDX  = (int*)grab((size_t)NUM_L * KTOP * 4);

  float* ZL = Z;
  float* ZR = Z + (size_t)NUM_L * CDIM;

  const dim3 blk256(256);
  auto cdiv = [](int a, int b) { return (a + b - 1) / b; };

  // --- GraphGenerator: zl = node_l@Wg, zr = node_r@Wg, topK(zl@zr^T) ---
  gemm_wmma_f32<<<dim3(CDIM / BN, NUM_L / BM), 256, 0, stream>>>(l_feat, Wg, ZL, CDIM, CDIM);
  gemm_wmma_f32<<<dim3(CDIM / BN, NUM_L / BM), 256, 0, stream>>>(r_feat, Wg, ZR, CDIM, CDIM);
  sim_topk_kernel<<<NUM_L / 16, 32, 0, stream>>>(ZL, ZR, IDX);

  // x = concat(node_l, node_r)
  concat_kernel<<<cdiv(NNODES * CDIM, 256), blk256, 0, stream>>>(l_feat, r_feat, X);
  build_edges1<<<cdiv(E1, 256), blk256, 0, stream>>>(IDX, SRC, DST);

  for (int g = 0; g < 2; ++g) {
    const int E = (g == 0) ? E1 : E2;
    if (g == 1) build_edges2<<<cdiv(E2, 256), blk256, 0, stream>>>(SRC, DST);
    for (int li = 0; li < 4; ++li) {
      // z = x @ W[li]
      gemm_wmma_f32<<<dim3(CDIM / BN, NNODES / BM), 256, 0, stream>>>(
          X, Ws + (size_t)li * CDIM * CDIM, Z, CDIM, CDIM);
      attn_scores<<<cdiv(NNODES * HEADS, 256), blk256, 0, stream>>>(
          Z, a_src + li * HEADS * HDIM, a_dst + li * HEADS * HDIM, EL, ER);
      fill_kernel<<<cdiv(NNODES * HEADS, 256), blk256, 0, stream>>>(MX, -3.402823466e38f, NNODES * HEADS);
      fill_kernel<<<cdiv(NNODES * HEADS, 256), blk256, 0, stream>>>(SX, 0.f, NNODES * HEADS);
      fill_kernel<<<cdiv(NNODES * CDIM, 256), blk256, 0, stream>>>(Y, 0.f, NNODES * CDIM);
      edge_logits_max<<<cdiv(E * HEADS, 256), blk256, 0, stream>>>(SRC, DST, EL, ER, EV, MX, E);
      edge_exp_sum<<<cdiv(E * HEADS, 256), blk256, 0, stream>>>(DST, EV, MX, SX, E);
      edge_scatter<<<cdiv(E * HEADS, 256), blk256, 0, stream>>>(SRC, DST, EV, SX, Z, Y, E);
      finalize_kernel<<<cdiv(NNODES * CDIM, 256), blk256, 0, stream>>>(
          Y, bias + li * CDIM, X, (li < 3) ? 1 : 0);
    }
  }
  gather_out<<<cdiv(32 * CDIM, 256), blk256, 0, stream>>>(X, out);
}